// CuGraphGATConv_32547262169649
// MI455X (gfx1250) — compile-verified
//
#include <hip/hip_runtime.h>

#define IN_C     128
#define HC       128   // HEADS * OUT_C
#define HEADS    4
#define DEG_MAX  8
#define NEG_SLOPE 0.2f

typedef __attribute__((ext_vector_type(2))) float v2f;
typedef __attribute__((ext_vector_type(8))) float v8f;

// ---------------------------------------------------------------------------
// Kernel 1: xp = x @ W  with V_WMMA_F32_16X16X4_F32 (exact f32 MAC path).
// One wave per 16-row x 128-col output strip: 8 accumulator tiles (64 VGPRs).
// Per K-step (K += 4): one A global_load_b64 + 8 LDS b64 reads feed 8 WMMAs.
// W (64 KB) is staged once per block into LDS, row-pair interleaved so each
// lane's {w[k][c], w[k+1][c]} pair is one contiguous ds_load_b64.
//
// ISA layouts (7.12.2):
//   A 16x4 f32 : lanes 0-15 M=lane {K,K+1}; lanes 16-31 M=lane-16 {K+2,K+3}
//   C 16x16 f32: VGPR r -> row r (lanes 0-15) / row r+8 (lanes 16-31)
// ---------------------------------------------------------------------------
__global__ __launch_bounds__(256) void gat_gemm_wmma(
    const float* __restrict__ x, const float* __restrict__ w,
    float* __restrict__ xp, int N, int row_tiles)
{
  __shared__ float wlds[IN_C * HC];          // 64 KB of the 320 KB WGP LDS
  // Stage W: element (r,c) -> wlds[((r>>1)*HC + c)*2 + (r&1)]
  for (int i = threadIdx.x; i < IN_C * HC; i += 256) {
    const int r = i >> 7;                    // row of W (K index)
    const int c = i & (HC - 1);              // col of W (output channel)
    wlds[((((r >> 1) << 7) | c) << 1) | (r & 1)] = w[i];
  }
  __syncthreads();

  const int lane = threadIdx.x & 31;
  const int wave = threadIdx.x >> 5;
  const int rt = blockIdx.x * 8 + wave;      // 16-row strip index
  if (rt >= row_tiles) return;

  const int half = lane >> 4;                // 0: lanes 0-15, 1: lanes 16-31
  const int l16  = lane & 15;
  int row = rt * 16 + l16;
  const bool valid = row < N;
  if (!valid) row = N - 1;                   // clamp: loads unconditional,
                                             // garbage rows never stored
  const float* xr = x + (size_t)row * IN_C;

  v8f c[8] = {};                             // 8 col tiles x 16x16
  #pragma unroll
  for (int k = 0; k < IN_C; k += 4) {
    const int ka = k + (half << 1);          // this half-wave's K pair
    const v2f a = *(const v2f*)(xr + ka);    // {x[row][ka], x[row][ka+1]}
    // LDS row-pair group for (ka, ka+1): pair row = k/2 + half
    const float* bb = wlds + ((((k >> 1) + half) << 7) << 1);
    #pragma unroll
    for (int ct = 0; ct < 8; ++ct) {
      const v2f b = *(const v2f*)(bb + ((ct * 16 + l16) << 1));
      c[ct] = __builtin_amdgcn_wmma_f32_16x16x4_f32(
          false, a, false, b, (short)0, c[ct], false, false);
    }
  }

  #pragma unroll
  for (int ct = 0; ct < 8; ++ct) {
    const int coln = ct * 16 + l16;
    #pragma unroll
    for (int r = 0; r < 8; ++r) {
      const int orow = rt * 16 + r + (half << 3);
      if (orow < N) xp[(size_t)orow * HC + coln] = c[ct][r];
    }
  }
}

// ---------------------------------------------------------------------------
// Kernel 2: a_src[n,h] = dot(xp[n, h*32 : h*32+32], att_src[h]),  same a_dst.
// One wave per node; lane holds 4 features (float4); 8 lanes per head reduce.
// ---------------------------------------------------------------------------
__global__ __launch_bounds__(256) void gat_alpha_terms(
    const float* __restrict__ xp, const float* __restrict__ att,
    float* __restrict__ a_src, float* __restrict__ a_dst, int N)
{
  const int lane = threadIdx.x & 31;
  const int wave = threadIdx.x >> 5;
  const int node = blockIdx.x * 8 + wave;
  if (node >= N) return;

  const float4 v  = *(const float4*)(xp + (size_t)node * HC + lane * 4);
  const float4 as = *(const float4*)(att + lane * 4);          // att_src slice
  const float4 ad = *(const float4*)(att + HC + lane * 4);     // att_dst slice
  float ps = v.x * as.x + v.y * as.y + v.z * as.z + v.w * as.w;
  float pd = v.x * ad.x + v.y * ad.y + v.z * ad.z + v.w * ad.w;

  // reduce over the 8 lanes that share head h = lane>>3
  #pragma unroll
  for (int off = 1; off < 8; off <<= 1) {
    ps += __shfl_xor(ps, off, 32);
    pd += __shfl_xor(pd, off, 32);
  }
  if ((lane & 7) == 0) {
    const int h = lane >> 3;
    a_src[(size_t)node * HEADS + h] = ps;
    a_dst[(size_t)node * HEADS + h] = pd;
  }
}

// ---------------------------------------------------------------------------
// Kernel 3: per-destination softmax over <=8 incoming edges + weighted
// aggregation. One wave per destination node. xp (51 MB) is L2-resident,
// so the random row gathers run at L2 bandwidth, not HBM.
// Phase A (attention layout): lane = 4*e + h -> logit(e,h); softmax via
// shfl_xor 4/8/16 (preserves h = lane&3).
// Phase B (feature layout): lane -> features 4*lane..4*lane+3 (head lane>>3);
// per edge: coalesced 512 B float4 row gather, weight broadcast by shfl.
// ---------------------------------------------------------------------------
__global__ __launch_bounds__(256) void gat_softmax_agg(
    const float* __restrict__ xp,
    const long long* __restrict__ row_ptr,
    const long long* __restrict__ col_ind,
    const float* __restrict__ a_src, const float* __restrict__ a_dst,
    const float* __restrict__ bias, float* __restrict__ out, int N)
{
  const int lane = threadIdx.x & 31;
  const int wave = threadIdx.x >> 5;
  const int node = blockIdx.x * 8 + wave;
  if (node >= N) return;

  const long long beg = row_ptr[node];
  const long long end = row_ptr[node + 1];
  int deg = (int)(end - beg);
  if (deg > DEG_MAX) deg = DEG_MAX;

  // ---- Phase A: logits + softmax, lane = 4*e + h ----
  const int e = lane >> 2;
  const int h = lane & 3;
  const bool evalid = e < deg;
  int srcE = 0;
  float logit = -3.0e38f;
  if (evalid) {
    srcE = (int)col_ind[beg + e];
    const float s = a_src[(size_t)srcE * HEADS + h] +
                    a_dst[(size_t)node * HEADS + h];
    logit = (s > 0.f) ? s : NEG_SLOPE * s;     // leaky_relu
  }
  float m = logit;
  m = fmaxf(m, __shfl_xor(m, 4, 32));
  m = fmaxf(m, __shfl_xor(m, 8, 32));
  m = fmaxf(m, __shfl_xor(m, 16, 32));
  float ex = evalid ? __expf(logit - m) : 0.f;
  float denom = ex;
  denom += __shfl_xor(denom, 4, 32);
  denom += __shfl_xor(denom, 8, 32);
  denom += __shfl_xor(denom, 16, 32);
  const float alpha = (denom > 0.f) ? (ex / denom) : 0.f;

  // ---- Phase B: aggregation, lane -> 4 features, head hf = lane>>3 ----
  const int hf = lane >> 3;
  float4 acc; acc.x = 0.f; acc.y = 0.f; acc.z = 0.f; acc.w = 0.f;
  for (int ee = 0; ee < deg; ++ee) {            // deg is wave-uniform
    const int s     = __shfl(srcE,  ee << 2, 32);          // edge ee's source
    const float wgt = __shfl(alpha, (ee << 2) | hf, 32);   // alpha(ee, hf)
    const float4 v  = *(const float4*)(xp + (size_t)s * HC + lane * 4);
    acc.x += wgt * v.x; acc.y += wgt * v.y;
    acc.z += wgt * v.z; acc.w += wgt * v.w;
  }
  const float4 b = *(const float4*)(bias + lane * 4);
  acc.x += b.x; acc.y += b.y; acc.z += b.z; acc.w += b.w;
  *(float4*)(out + (size_t)node * HC + lane * 4) = acc;
}

// ---------------------------------------------------------------------------
// Launch. Inputs (setup_inputs order):
//   0: x [N,128] f32    1: csr_row_ptr [N+1] i64   2: csr_col_ind [E] i64
//   3: max_num_neighbors (scalar)  4: lin_w [128,128] f32
//   5: att [256] f32    6: bias [128] f32
// Workspace: xp [N*128] f32, a_src [N*4] f32, a_dst [N*4] f32  (~54.4 MB)
// ---------------------------------------------------------------------------
extern "C" void kernel_launch(void* const* d_in, const int* in_sizes, int n_in,
                              void* d_out, int out_size, void* d_ws, size_t ws_size,
                              hipStream_t stream)
{
  const float*     x       = (const float*)d_in[0];
  const long long* row_ptr = (const long long*)d_in[1];
  const long long* col_ind = (const long long*)d_in[2];
  const float*     lin_w   = (const float*)d_in[4];
  const float*     att     = (const float*)d_in[5];
  const float*     bias    = (const float*)d_in[6];
  const int N = in_sizes[0] / IN_C;

  float* xp    = (float*)d_ws;
  float* a_src = xp + (size_t)N * HC;
  float* a_dst = a_src + (size_t)N * HEADS;

  const int row_tiles = (N + 15) / 16;            // 16-row strips
  hipLaunchKernelGGL(gat_gemm_wmma, dim3((row_tiles + 7) / 8), dim3(256), 0,
                     stream, x, lin_w, xp, N, row_tiles);

  const int blocks = (N + 7) / 8;                 // 8 waves (nodes) per block
  hipLaunchKernelGGL(gat_alpha_terms, dim3(blocks), dim3(256), 0, stream,
                     xp, att, a_src, a_dst, N);
  hipLaunchKernelGGL(gat_softmax_agg, dim3(blocks), dim3(256), 0, stream,
                     xp, row_ptr, col_ind, a_src, a_dst, bias,
                     (float*)d_out, N);
}